// OptimalTransport_78082505441398
// MI455X (gfx1250) — compile-verified
//
#include <hip/hip_runtime.h>
#include <math.h>

// ---------------- problem constants (from reference) ----------------
constexpr int BATCH = 8;
constexpr int L     = 2048;
constexpr int S     = 2048;
constexpr int C     = 256;
constexpr int NP1   = 2049;          // L+1 == S+1
constexpr int ITERS = 20;

// norm = -log(m+n) = -log(4096);  log(2048)
__device__ __constant__ float d_dummy; // keep compiler happy if unused
#define NORMC   (-8.31776616671934f)   // -log(4096)
#define LOGNC   ( 7.62461898616594f)   //  log(2048)

constexpr int LCH = 128;             // rows per column-pass chunk
constexpr int NCH = (NP1 + LCH - 1) / LCH;   // 17

typedef __attribute__((ext_vector_type(2))) float v2f;
typedef __attribute__((ext_vector_type(8))) float v8f;

// ---------------- 1) LDS-tiled transpose  [b][rows][cols] -> [b][cols][rows] ----
__global__ void transpose_k(const float* __restrict__ in, float* __restrict__ out,
                            int rows, int cols) {
  __shared__ float tile[32][33];
  const int b  = blockIdx.z;
  const float* ib = in  + (size_t)b * rows * cols;
  float*       ob = out + (size_t)b * rows * cols;
  const int c0 = blockIdx.x * 32;
  const int r0 = blockIdx.y * 32;
  const int tx = threadIdx.x, ty = threadIdx.y;       // block (32,8)
#pragma unroll
  for (int i = 0; i < 32; i += 8)
    tile[ty + i][tx] = ib[(size_t)(r0 + ty + i) * cols + (c0 + tx)];
  __syncthreads();
#pragma unroll
  for (int i = 0; i < 32; i += 8)
    ob[(size_t)(c0 + ty + i) * rows + (r0 + tx)] = tile[tx][ty + i];
}

// ---------------- 2) WMMA f32 GEMM: Z[b][l][s] = sum_c f0T[b][c][l]*f1T[b][c][s] --
// grid.x = BATCH * (L/16) * 4 ; block = 128 (4 waves); wave -> 16(M) x 128(N) strip
__global__ void gemm_wmma_k(const float* __restrict__ f0T,
                            const float* __restrict__ f1T,
                            float* __restrict__ Z) {
  const int wave = threadIdx.x >> 5;
  const int lane = threadIdx.x & 31;
  const int mrow = lane & 15;
  const int half = lane >> 4;

  int gid   = blockIdx.x;
  const int b     = gid / (128 * 4);
  const int rem   = gid % (128 * 4);
  const int mtile = rem >> 2;             // 0..127
  const int nblk  = rem & 3;              // 0..3
  const int l0 = mtile * 16;
  const int s0 = nblk * 512 + wave * 128;

  const float* A  = f0T + (size_t)b * C * L;   // [C][L]
  const float* Bm = f1T + (size_t)b * C * S;   // [C][S]

  v8f acc[8] = {};
  for (int k0 = 0; k0 < C; k0 += 4) {
    const int ka = k0 + 2 * half;               // lanes 0-15: k0,k0+1 ; 16-31: k0+2,k0+3
    v2f a;
    a.x = A[(size_t)ka       * L + (l0 + mrow)];
    a.y = A[(size_t)(ka + 1) * L + (l0 + mrow)];
#pragma unroll
    for (int t = 0; t < 8; ++t) {
      v2f bb;
      bb.x = Bm[(size_t)ka       * S + (s0 + 16 * t + mrow)];
      bb.y = Bm[(size_t)(ka + 1) * S + (s0 + 16 * t + mrow)];
      acc[t] = __builtin_amdgcn_wmma_f32_16x16x4_f32(
                 /*neg_a=*/false, a, /*neg_b=*/false, bb,
                 /*c_mod=*/(short)0, acc[t], /*reuse_a=*/false, /*reuse_b=*/false);
    }
  }
  // C/D layout: vgpr r, lane -> row = r + 8*half, col = mrow
  float* zb = Z + (size_t)b * NP1 * NP1;
#pragma unroll
  for (int t = 0; t < 8; ++t) {
#pragma unroll
    for (int r = 0; r < 8; ++r) {
      const int row = l0 + r + 8 * half;
      const int col = s0 + 16 * t + mrow;
      zb[(size_t)row * NP1 + col] = acc[t][r];
    }
  }
}

// ---------------- 3) fill dustbin row / col with bin_score ----------------
__global__ void fill_bins_k(float* __restrict__ Z, const float* __restrict__ binp) {
  const float bin = *binp;
  const int b = blockIdx.y;
  float* zb = Z + (size_t)b * NP1 * NP1;
  const int i = blockIdx.x * 256 + threadIdx.x;
  if (i < NP1) {
    zb[(size_t)i * NP1 + (NP1 - 1)]   = bin;   // last column
    zb[(size_t)(NP1 - 1) * NP1 + i]   = bin;   // last row (corner covered twice, same value)
  }
}

// ---------------- 4) zero u, v ----------------
__global__ void init_uv_k(float* __restrict__ u, float* __restrict__ v) {
  const int i = blockIdx.x * 256 + threadIdx.x;
  if (i < BATCH * NP1) { u[i] = 0.f; v[i] = 0.f; }
}

// ---------------- 5) row pass: u = log_mu - lse_s(Z + v) ----------------
__global__ void row_pass_k(const float* __restrict__ Z, const float* __restrict__ v,
                           float* __restrict__ u) {
  const int l = blockIdx.x;
  const int b = blockIdx.y;
  const float* row = Z + ((size_t)b * NP1 + l) * NP1;
  const float* vb  = v + (size_t)b * NP1;
  const int tid = threadIdx.x;

  float m = -3.0e38f, s = 0.f;
  for (int j = tid; j < NP1; j += 256) {
    const float x = row[j] + vb[j];
    if (x > m) { s = s * __expf(m - x) + 1.f; m = x; }
    else       { s += __expf(x - m); }
  }
  __shared__ float sm[256], ss[256];
  sm[tid] = m; ss[tid] = s;
  __syncthreads();
  for (int w = 128; w > 0; w >>= 1) {
    if (tid < w) {
      const float m2 = sm[tid + w], s2 = ss[tid + w];
      const float M  = fmaxf(sm[tid], m2);
      ss[tid] = ss[tid] * __expf(sm[tid] - M) + s2 * __expf(m2 - M);
      sm[tid] = M;
    }
    __syncthreads();
  }
  if (tid == 0) {
    const float lse = sm[0] + __logf(ss[0]);
    const float lmu = (l < NP1 - 1) ? NORMC : (LOGNC + NORMC);
    u[(size_t)b * NP1 + l] = lmu - lse;
  }
}

// ---------------- 6a) column pass, partial chunks ----------------
__global__ void col_partial_k(const float* __restrict__ Z, const float* __restrict__ u,
                              float* __restrict__ pm, float* __restrict__ ps) {
  const int b  = blockIdx.z;
  const int s  = blockIdx.x * 256 + threadIdx.x;
  const int ch = blockIdx.y;
  if (s >= NP1) return;
  const int l0 = ch * LCH;
  const int l1 = (l0 + LCH < NP1) ? (l0 + LCH) : NP1;
  const float* ub = u + (size_t)b * NP1;
  const float* zb = Z + (size_t)b * NP1 * NP1;

  float m = -3.0e38f, acc = 0.f;
  for (int l = l0; l < l1; ++l) {
    const float x = zb[(size_t)l * NP1 + s] + ub[l];
    if (x > m) { acc = acc * __expf(m - x) + 1.f; m = x; }
    else       { acc += __expf(x - m); }
  }
  const size_t pidx = ((size_t)b * NCH + ch) * NP1 + s;
  pm[pidx] = m; ps[pidx] = acc;
}

// ---------------- 6b) column pass, combine -> v ----------------
__global__ void col_final_k(const float* __restrict__ pm, const float* __restrict__ ps,
                            float* __restrict__ v) {
  const int b = blockIdx.y;
  const int s = blockIdx.x * 256 + threadIdx.x;
  if (s >= NP1) return;
  float m = -3.0e38f, acc = 0.f;
  for (int ch = 0; ch < NCH; ++ch) {
    const size_t pidx = ((size_t)b * NCH + ch) * NP1 + s;
    const float m2 = pm[pidx], s2 = ps[pidx];
    const float M  = fmaxf(m, m2);
    acc = acc * __expf(m - M) + s2 * __expf(m2 - M);
    m = M;
  }
  const float lse = m + __logf(acc);
  const float lnu = (s < NP1 - 1) ? NORMC : (LOGNC + NORMC);
  v[(size_t)b * NP1 + s] = lnu - lse;
}

// ---------------- 7) final: Z += u + v - norm (in place) ----------------
__global__ void final_add_k(float* __restrict__ Z, const float* __restrict__ u,
                            const float* __restrict__ v) {
  const int i = blockIdx.x;
  const int b = blockIdx.y;
  float*       row = Z + ((size_t)b * NP1 + i) * NP1;
  const float* vb  = v + (size_t)b * NP1;
  const float  ui  = u[(size_t)b * NP1 + i] - NORMC;   // subtracting norm == adding -norm
  for (int j = threadIdx.x; j < NP1; j += 256)
    row[j] += ui + vb[j];
}

// ---------------- launcher ----------------
extern "C" void kernel_launch(void* const* d_in, const int* in_sizes, int n_in,
                              void* d_out, int out_size, void* d_ws, size_t ws_size,
                              hipStream_t stream) {
  (void)in_sizes; (void)n_in; (void)out_size; (void)ws_size;
  const float* f0  = (const float*)d_in[0];   // [B][L][C] f32
  const float* f1  = (const float*)d_in[1];   // [B][S][C] f32
  const float* bin = (const float*)d_in[2];   // scalar f32
  float* Z = (float*)d_out;                   // [B][NP1][NP1] f32

  // workspace layout (~36 MB)
  float* f0T = (float*)d_ws;                                // B*C*L
  float* f1T = f0T + (size_t)BATCH * C * L;                 // B*C*S
  float* u   = f1T + (size_t)BATCH * C * S;                 // B*NP1
  float* v   = u   + (size_t)BATCH * NP1;                   // B*NP1
  float* pm  = v   + (size_t)BATCH * NP1;                   // B*NCH*NP1
  float* ps  = pm  + (size_t)BATCH * NCH * NP1;             // B*NCH*NP1

  // 1) transpose features to [C][rows] for coalesced WMMA fragment loads
  {
    dim3 blk(32, 8);
    dim3 g0(C / 32, L / 32, BATCH);
    transpose_k<<<g0, blk, 0, stream>>>(f0, f0T, L, C);
    dim3 g1(C / 32, S / 32, BATCH);
    transpose_k<<<g1, blk, 0, stream>>>(f1, f1T, S, C);
  }

  // 2) GEMM into Z (scores region)
  gemm_wmma_k<<<dim3(BATCH * 128 * 4), dim3(128), 0, stream>>>(f0T, f1T, Z);

  // 3) dustbin row/col
  fill_bins_k<<<dim3((NP1 + 255) / 256, BATCH), dim3(256), 0, stream>>>(Z, bin);

  // 4) u = v = 0
  init_uv_k<<<dim3((BATCH * NP1 + 255) / 256), dim3(256), 0, stream>>>(u, v);

  // 5) 20 Sinkhorn iterations
  for (int it = 0; it < ITERS; ++it) {
    row_pass_k   <<<dim3(NP1, BATCH),                dim3(256), 0, stream>>>(Z, v, u);
    col_partial_k<<<dim3((NP1 + 255) / 256, NCH, BATCH), dim3(256), 0, stream>>>(Z, u, pm, ps);
    col_final_k  <<<dim3((NP1 + 255) / 256, BATCH),  dim3(256), 0, stream>>>(pm, ps, v);
  }

  // 6) Z + u + v - norm, in place
  final_add_k<<<dim3(NP1, BATCH), dim3(256), 0, stream>>>(Z, u, v);
}